// GCN_49503793053816
// MI455X (gfx1250) — compile-verified
//
#include <hip/hip_runtime.h>

#define N_NODES 100000
#define N_EDGES 1000000
#define D 64
#define LN_EPS 1e-5f

typedef __attribute__((ext_vector_type(2))) float v2f;
typedef __attribute__((ext_vector_type(8))) float v8f;

// ---------------------------------------------------------------- utilities
__global__ void zero_f32_kernel(float* __restrict__ p, int n) {
  int i = blockIdx.x * blockDim.x + threadIdx.x;
  if (i < n) p[i] = 0.0f;
}

// degree accumulation: deg buffers are pre-zeroed, one thread per edge
__global__ void degree_kernel(const int* __restrict__ src, const int* __restrict__ dst,
                              float* __restrict__ out_deg, float* __restrict__ in_deg, int e) {
  int i = blockIdx.x * blockDim.x + threadIdx.x;
  if (i < e) {
    atomicAdd(&out_deg[src[i]], 1.0f);
    atomicAdd(&in_deg[dst[i]], 1.0f);
  }
}

// in-place: deg -> rsqrt(max(deg,1))
__global__ void norm_kernel(float* __restrict__ out_norm, float* __restrict__ in_norm, int n) {
  int i = blockIdx.x * blockDim.x + threadIdx.x;
  if (i < n) {
    out_norm[i] = rsqrtf(fmaxf(out_norm[i], 1.0f));
    in_norm[i]  = rsqrtf(fmaxf(in_norm[i], 1.0f));
  }
}

// ------------------------------------------------ edge gather/scatter phase
// one wave (32 lanes) per edge; each lane moves 2 floats (float2 gather,
// 2x global_atomic_add_f32 scatter). out_norm scaling fused here.
__global__ void scatter_kernel(const float* __restrict__ h,
                               const float* __restrict__ out_norm,
                               const int* __restrict__ src, const int* __restrict__ dst,
                               float* __restrict__ agg, int e) {
  int g    = blockIdx.x * blockDim.x + threadIdx.x;
  int edge = g >> 5;
  int lane = g & 31;
  if (edge >= e) return;
  int s = src[edge];
  int d = dst[edge];
  float on = out_norm[s];
  float2 v = *(reinterpret_cast<const float2*>(h + (size_t)s * D) + lane);
  float* ap = agg + (size_t)d * D + lane * 2;
  atomicAdd(ap,     v.x * on);
  atomicAdd(ap + 1, v.y * on);
}

// ------------------------------------------------ fused WMMA layer kernel
// per wave: 16 nodes.  H_new = (agg*in_norm)@Wc + h@Wr + (bc+br), both GEMMs
// accumulated into one f32 WMMA accumulator chain, then LN+ReLU+JK via LDS.
__global__ __launch_bounds__(256) void gcn_layer_kernel(
    const float* __restrict__ agg, const float* __restrict__ in_norm,
    const float* __restrict__ h_in,
    const float* __restrict__ Wc, const float* __restrict__ bc,
    const float* __restrict__ Wr, const float* __restrict__ br,
    const float* __restrict__ lng, const float* __restrict__ lnb,
    float* __restrict__ h_out, float* __restrict__ h_final, int n)
{
  __shared__ float lds[8][16 * D];
  const int wave = threadIdx.x >> 5;
  const int lane = threadIdx.x & 31;
  const int row  = lane & 15;
  const int kb   = (lane >> 4) * 2;              // 0 or 2
  const int base = (blockIdx.x * 8 + wave) * 16; // first node of this wave's tile
  int node  = base + row;
  int cnode = node < n ? node : (n - 1);         // clamp loads; stores are masked

  // preload A fragments: conv path (agg * in_norm) and residual path (h)
  v2f aC[16], aR[16];
  const float inrm = in_norm[cnode];
  const float* arow = agg  + (size_t)cnode * D;
  const float* hrow = h_in + (size_t)cnode * D;
#pragma unroll
  for (int kt = 0; kt < 16; ++kt) {
    int k0 = kt * 4 + kb;
    aC[kt][0] = arow[k0]     * inrm;
    aC[kt][1] = arow[k0 + 1] * inrm;
    aR[kt][0] = hrow[k0];
    aR[kt][1] = hrow[k0 + 1];
  }

  const int ncol = lane & 15;
#pragma unroll
  for (int nt = 0; nt < 4; ++nt) {
    v8f c = {};
    const int col = nt * 16 + ncol;
#pragma unroll
    for (int kt = 0; kt < 16; ++kt) {
      int kr = kt * 4 + kb;
      v2f bC, bR;
      bC[0] = Wc[kr * D + col];
      bC[1] = Wc[(kr + 1) * D + col];
      bR[0] = Wr[kr * D + col];
      bR[1] = Wr[(kr + 1) * D + col];
      c = __builtin_amdgcn_wmma_f32_16x16x4_f32(false, aC[kt], false, bC, (short)0, c, false, false);
      c = __builtin_amdgcn_wmma_f32_16x16x4_f32(false, aR[kt], false, bR, (short)0, c, false, false);
    }
    const float bias = bc[col] + br[col];
    const int m0 = (lane < 16) ? 0 : 8;
#pragma unroll
    for (int r = 0; r < 8; ++r)
      lds[wave][(m0 + r) * D + col] = c[r] + bias;
  }
  __syncthreads();

  // LayerNorm + ReLU + JK accumulate. 2 lanes per row, 32 elements each.
  const int rr = lane >> 1;
  const int j0 = (lane & 1) * 32;
  const float* myrow = &lds[wave][rr * D];
  float s = 0.0f, sq = 0.0f;
#pragma unroll
  for (int j = 0; j < 32; ++j) {
    float x = myrow[j0 + j];
    s  += x;
    sq += x * x;
  }
  s  += __shfl_xor(s, 1);
  sq += __shfl_xor(sq, 1);
  const float mu   = s * (1.0f / 64.0f);
  const float var  = sq * (1.0f / 64.0f) - mu * mu;
  const float rstd = rsqrtf(var + LN_EPS);
  const int onode = base + rr;
  if (onode < n) {
    float* ho = h_out   + (size_t)onode * D;
    float* hf = h_final + (size_t)onode * D;
#pragma unroll
    for (int j = 0; j < 32; ++j) {
      int jj = j0 + j;
      float y = (myrow[jj] - mu) * rstd * lng[jj] + lnb[jj];
      y = fmaxf(y, 0.0f);
      ho[jj] = y;
      hf[jj] += y;
    }
  }
}

// ------------------------------------------------ final prediction GEMM
__global__ __launch_bounds__(256) void pred_kernel(
    const float* __restrict__ hf, const float* __restrict__ W,
    const float* __restrict__ b, float* __restrict__ out, int n)
{
  const int wave = threadIdx.x >> 5;
  const int lane = threadIdx.x & 31;
  const int row  = lane & 15;
  const int kb   = (lane >> 4) * 2;
  const int base = (blockIdx.x * 8 + wave) * 16;
  int node  = base + row;
  int cnode = node < n ? node : (n - 1);

  v2f a[16];
  const float* ar = hf + (size_t)cnode * D;
#pragma unroll
  for (int kt = 0; kt < 16; ++kt) {
    int k0 = kt * 4 + kb;
    a[kt][0] = ar[k0];
    a[kt][1] = ar[k0 + 1];
  }

  const int ncol = lane & 15;
#pragma unroll
  for (int nt = 0; nt < 4; ++nt) {
    v8f c = {};
    const int col = nt * 16 + ncol;
#pragma unroll
    for (int kt = 0; kt < 16; ++kt) {
      int kr = kt * 4 + kb;
      v2f bf;
      bf[0] = W[kr * D + col];
      bf[1] = W[(kr + 1) * D + col];
      c = __builtin_amdgcn_wmma_f32_16x16x4_f32(false, a[kt], false, bf, (short)0, c, false, false);
    }
    const float bias = b[col];
    const int m0 = (lane < 16) ? 0 : 8;
#pragma unroll
    for (int r = 0; r < 8; ++r) {
      int m = base + m0 + r;
      if (m < n) out[(size_t)m * D + col] = c[r] + bias;
    }
  }
}

// ---------------------------------------------------------------- launcher
extern "C" void kernel_launch(void* const* d_in, const int* in_sizes, int n_in,
                              void* d_out, int out_size, void* d_ws, size_t ws_size,
                              hipStream_t stream) {
  (void)in_sizes; (void)n_in; (void)out_size; (void)ws_size;

  const float* feats  = (const float*)d_in[0];
  const int*   src    = (const int*)  d_in[1];
  const int*   dst    = (const int*)  d_in[2];
  const float* conv_W = (const float*)d_in[3];
  const float* conv_b = (const float*)d_in[4];
  const float* res_W  = (const float*)d_in[5];
  const float* res_b  = (const float*)d_in[6];
  const float* ln_g   = (const float*)d_in[7];
  const float* ln_b   = (const float*)d_in[8];
  const float* pred_W = (const float*)d_in[9];
  const float* pred_b = (const float*)d_in[10];
  float* out = (float*)d_out;

  char* ws = (char*)d_ws;
  size_t off = 0;
  auto alloc = [&](size_t bytes) -> float* {
    float* p = (float*)(ws + off);
    off += (bytes + 255) & ~(size_t)255;
    return p;
  };
  const size_t nodeF = (size_t)N_NODES * 4;
  const size_t featF = (size_t)N_NODES * D * 4;
  float* out_norm = alloc(nodeF);
  float* in_norm  = alloc(nodeF);
  float* agg      = alloc(featF);
  float* hA       = alloc(featF);
  float* hB       = alloc(featF);
  float* h_final  = alloc(featF);

  const int TB = 256;
  const int nFeat   = N_NODES * D;
  const int gFeat   = (nFeat + TB - 1) / TB;
  const int gNode   = (N_NODES + TB - 1) / TB;
  const int gEdge   = (N_EDGES + TB - 1) / TB;
  const int gScat   = ((N_EDGES * 32) + TB - 1) / TB;
  const int gTile   = (N_NODES + 127) / 128;   // 8 waves * 16 nodes per block

  // degrees -> norms
  zero_f32_kernel<<<gNode, TB, 0, stream>>>(out_norm, N_NODES);
  zero_f32_kernel<<<gNode, TB, 0, stream>>>(in_norm,  N_NODES);
  degree_kernel<<<gEdge, TB, 0, stream>>>(src, dst, out_norm, in_norm, N_EDGES);
  norm_kernel<<<gNode, TB, 0, stream>>>(out_norm, in_norm, N_NODES);

  // JK accumulator
  zero_f32_kernel<<<gFeat, TB, 0, stream>>>(h_final, nFeat);

  const float* h_in = feats;
  float* bufs[2] = { hA, hB };
  for (int layer = 0; layer < 3; ++layer) {
    float* h_out = bufs[layer & 1];
    zero_f32_kernel<<<gFeat, TB, 0, stream>>>(agg, nFeat);
    scatter_kernel<<<gScat, TB, 0, stream>>>(h_in, out_norm, src, dst, agg, N_EDGES);
    gcn_layer_kernel<<<gTile, TB, 0, stream>>>(
        agg, in_norm, h_in,
        conv_W + (size_t)layer * D * D, conv_b + (size_t)layer * D,
        res_W  + (size_t)layer * D * D, res_b  + (size_t)layer * D,
        ln_g   + (size_t)layer * D,     ln_b   + (size_t)layer * D,
        h_out, h_final, N_NODES);
    h_in = h_out;
  }

  pred_kernel<<<gTile, TB, 0, stream>>>(h_final, pred_W, pred_b, out, N_NODES);
}